// CoxLoss_15668040695880
// MI455X (gfx1250) — compile-verified
//
#include <hip/hip_runtime.h>
#include <stdint.h>

#define MBITS      23
#define MBUCK      (1u << MBITS)           // 8,388,608 buckets (one per representable time)
#define SCAN_TPB   256
#define SCAN_ITEMS 16
#define SCAN_CHUNK (SCAN_TPB * SCAN_ITEMS) // 4096
#define SCAN_BLOCKS ((int)(MBUCK / SCAN_CHUNK)) // 2048
#define TDM_TILE   8192                    // floats per block staged through LDS by TDM (32KB)
#define LOSS_TPB   256
#define LOSS_ITEMS 8

typedef unsigned __attribute__((ext_vector_type(4))) uint4v;
typedef int      __attribute__((ext_vector_type(4))) int4v;
typedef int      __attribute__((ext_vector_type(8))) int8v;

__device__ __forceinline__ unsigned bucket_of(float t) {
  // t = k * 2^-23  ->  t * 2^23 is exact; monotone key, clamped defensively.
  int b = (int)(t * (float)MBUCK);
  b = b < 0 ? 0 : b;
  b = b > (int)(MBUCK - 1u) ? (int)(MBUCK - 1u) : b;
  return (unsigned)b;
}

// ---------------------------------------------------------------------------
// K1: bucket histogram. Streaming load of `times` staged global->LDS via the
// Tensor Data Mover (async tensor op, TENSORcnt), then counted with u32 atomics.
// ---------------------------------------------------------------------------
__global__ void k_count(const float* __restrict__ times, unsigned* __restrict__ cnt, int n) {
  __shared__ float sh[TDM_TILE];
  int start = (int)blockIdx.x * TDM_TILE;
  int L = n - start;
  if (L > TDM_TILE) L = TDM_TILE;
  if (L <= 0) return;  // uniform per block
#if defined(__HIP_DEVICE_COMPILE__) && __has_builtin(__builtin_amdgcn_tensor_load_to_lds)
  if (threadIdx.x < 32) {  // one wave issues the TDM op (EXEC-independent DMA)
    unsigned loff =
        (unsigned)(unsigned long long)(__attribute__((address_space(3))) void*)(void*)sh;
    unsigned long long ga = (unsigned long long)(const void*)(times + start);
    unsigned ne = (unsigned)L;
    // D# group0: count=1 | lds_addr | global_addr[56:0] | type=2 ("image")
    uint4v g0;
    g0.x = 1u;
    g0.y = loff;
    g0.z = (unsigned)(ga & 0xFFFFFFFFull);
    g0.w = (unsigned)((ga >> 32) & 0x1FFFFFFull) | (2u << 30);
    // D# group1: wg_mask=0, data_size=2 (4B), 1-D tile: tensor_dim0=tile_dim0=ne,
    // tensor_dim1=1, tile_dim1=1, strides = ne.
    int8v g1;
    g1[0] = (int)(2u << 16);
    g1[1] = (int)((ne & 0xFFFFu) << 16);                            // tensor_dim0 lo16
    g1[2] = (int)((ne >> 16) | (1u << 16));                         // dim0 hi16 | dim1 lo16=1
    g1[3] = (int)((ne & 0xFFFFu) << 16);                            // dim1 hi16=0 | tile_dim0
    g1[4] = 1;                                                      // tile_dim1=1, tile_dim2=0
    g1[5] = (int)ne;                                                // dim0_stride lo32
    g1[6] = (int)(((ne >> 16) & 0xFFFFu) | ((ne & 0xFFFFu) << 16)); // stride0 hi | stride1 lo
    g1[7] = (int)(ne >> 16);                                        // stride1 hi
    int4v gz4 = {0, 0, 0, 0};
    int8v gz8 = {0, 0, 0, 0, 0, 0, 0, 0};
    __builtin_amdgcn_tensor_load_to_lds(g0, g1, gz4, gz4, gz8, 0);
    __builtin_amdgcn_s_wait_tensorcnt(0);
  }
#else
  for (int i = (int)threadIdx.x; i < L; i += (int)blockDim.x) sh[i] = times[start + i];
#endif
  __syncthreads();
  for (int i = (int)threadIdx.x; i < L; i += (int)blockDim.x) {
    atomicAdd(&cnt[bucket_of(sh[i])], 1u);
  }
}

// ---------------------------------------------------------------------------
// Suffix-exclusive scan over the MBUCK bucket array (descending bucket order),
// in place, 3-phase hierarchical, fully deterministic. Works for u32 and f32.
// ---------------------------------------------------------------------------
template <typename T>
__global__ void k_scan_ph1(const T* __restrict__ in, T* __restrict__ bpart) {
  int base = (int)blockIdx.x * SCAN_CHUNK;
  T s = (T)0;
  for (int k = (int)threadIdx.x; k < SCAN_CHUNK; k += SCAN_TPB) {
    s += in[(int)MBUCK - 1 - (base + k)];
  }
  __shared__ T red[SCAN_TPB];
  red[threadIdx.x] = s;
  __syncthreads();
  for (int o = SCAN_TPB / 2; o > 0; o >>= 1) {
    if ((int)threadIdx.x < o) red[threadIdx.x] += red[threadIdx.x + o];
    __syncthreads();
  }
  if (threadIdx.x == 0) bpart[blockIdx.x] = red[0];
}

template <typename T>
__global__ void k_scan_ph2(T* __restrict__ bpart) {
  const int IT = SCAN_BLOCKS / SCAN_TPB;  // 8
  __shared__ T sums[SCAN_TPB];
  T loc[IT];
  T s = (T)0;
  for (int k = 0; k < IT; ++k) {
    loc[k] = bpart[(int)threadIdx.x * IT + k];
    s += loc[k];
  }
  sums[threadIdx.x] = s;
  __syncthreads();
  for (int o = 1; o < SCAN_TPB; o <<= 1) {
    T v = ((int)threadIdx.x >= o) ? sums[threadIdx.x - o] : (T)0;
    __syncthreads();
    sums[threadIdx.x] += v;
    __syncthreads();
  }
  T run = (threadIdx.x == 0) ? (T)0 : sums[threadIdx.x - 1];
  for (int k = 0; k < IT; ++k) {
    bpart[(int)threadIdx.x * IT + k] = run;
    run += loc[k];
  }
}

template <typename T>
__global__ void k_scan_ph3(T* __restrict__ data, const T* __restrict__ bpart) {
  int base = (int)blockIdx.x * SCAN_CHUNK;
  int tb = base + (int)threadIdx.x * SCAN_ITEMS;
  T loc[SCAN_ITEMS];
  T s = (T)0;
  for (int k = 0; k < SCAN_ITEMS; ++k) {
    loc[k] = data[(int)MBUCK - 1 - (tb + k)];
    s += loc[k];
  }
  __shared__ T sums[SCAN_TPB];
  sums[threadIdx.x] = s;
  __syncthreads();
  for (int o = 1; o < SCAN_TPB; o <<= 1) {
    T v = ((int)threadIdx.x >= o) ? sums[threadIdx.x - o] : (T)0;
    __syncthreads();
    sums[threadIdx.x] += v;
    __syncthreads();
  }
  T run = bpart[blockIdx.x] + ((threadIdx.x == 0) ? (T)0 : sums[threadIdx.x - 1]);
  for (int k = 0; k < SCAN_ITEMS; ++k) {
    T v = loc[k];
    data[(int)MBUCK - 1 - (tb + k)] = run;  // exclusive suffix sum
    run += v;
  }
}

// ---------------------------------------------------------------------------
// K3: counting-sort scatter of element indices into bucket segments.
// ---------------------------------------------------------------------------
__global__ void k_scatter(const float* __restrict__ times, const unsigned* __restrict__ off,
                          unsigned* __restrict__ pos, unsigned* __restrict__ sidx, int n) {
  int i = (int)blockIdx.x * (int)blockDim.x + (int)threadIdx.x;
  if (i >= n) return;
  unsigned b = bucket_of(times[i]);
  unsigned slot = off[b] + atomicAdd(&pos[b], 1u);
  sidx[slot] = (unsigned)i;
}

// ---------------------------------------------------------------------------
// K4: per-bucket sum of exp(log_risk), accumulated in ascending original-index
// order (deterministic regardless of scatter order). Bucket sizes ~Poisson(1).
// ---------------------------------------------------------------------------
__global__ void k_bucketsum(const unsigned* __restrict__ off, const unsigned* __restrict__ sidx,
                            const float* __restrict__ logr, float* __restrict__ S, int n) {
  unsigned b = (unsigned)blockIdx.x * blockDim.x + threadIdx.x;
  if (b >= MBUCK) return;
  unsigned s0 = off[b];
  unsigned s1 = (b == 0) ? (unsigned)n : off[b - 1];
  unsigned L = s1 - s0;
  float acc = 0.0f;
  unsigned last = 0;
  bool first = true;
  for (unsigned c = 0; c < L; ++c) {  // O(L^2) selection; E[L]=1
    unsigned best = 0xFFFFFFFFu;
    for (unsigned q = s0; q < s1; ++q) {
      unsigned j = sidx[q];
      if ((first || j > last) && j < best) best = j;
    }
    acc += expf(logr[best]);
    last = best;
    first = false;
  }
  S[b] = acc;
}

// ---------------------------------------------------------------------------
// K5: per-event loss. at_risk = T[b] + within-bucket ordered partial sum.
// Deterministic per-block partials; event count via u32 atomics.
// ---------------------------------------------------------------------------
__global__ void k_loss(const float* __restrict__ logr, const float* __restrict__ times,
                       const int* __restrict__ censor, const unsigned* __restrict__ off,
                       const unsigned* __restrict__ sidx, const float* __restrict__ Tsuf,
                       float* __restrict__ lpart, unsigned* __restrict__ nev, int n) {
  __shared__ float redf[LOSS_TPB];
  __shared__ unsigned redu[LOSS_TPB];
  int base = (int)blockIdx.x * (LOSS_TPB * LOSS_ITEMS);
  float acc = 0.0f;
  unsigned ev = 0;
  for (int k = 0; k < LOSS_ITEMS; ++k) {
    int i = base + (int)threadIdx.x + k * LOSS_TPB;
    if (i < n && censor[i] == 1) {
      float t = times[i];
      unsigned b = bucket_of(t);
      unsigned s0 = off[b];
      unsigned s1 = (b == 0) ? (unsigned)n : off[b - 1];
      unsigned L = s1 - s0;
      float w = 0.0f;
      unsigned last = 0;
      bool first = true;
      for (unsigned c = 0; c < L; ++c) {
        unsigned best = 0xFFFFFFFFu;
        for (unsigned q = s0; q < s1; ++q) {
          unsigned j = sidx[q];
          if ((first || j > last) && j < best) best = j;
        }
        last = best;
        first = false;
        float tj = times[best];
        if (tj > t || (tj == t && best <= (unsigned)i)) w += expf(logr[best]);
      }
      acc += logr[i] - logf(Tsuf[b] + w + 1e-15f);
      ev++;
    }
  }
  redf[threadIdx.x] = acc;
  redu[threadIdx.x] = ev;
  __syncthreads();
  for (int o = LOSS_TPB / 2; o > 0; o >>= 1) {
    if ((int)threadIdx.x < o) {
      redf[threadIdx.x] += redf[threadIdx.x + o];
      redu[threadIdx.x] += redu[threadIdx.x + o];
    }
    __syncthreads();
  }
  if (threadIdx.x == 0) {
    lpart[blockIdx.x] = redf[0];
    atomicAdd(nev, redu[0]);
  }
}

__global__ void k_final(const float* __restrict__ lpart, int nblocks,
                        const unsigned* __restrict__ nev, float* __restrict__ out) {
  __shared__ float red[256];
  float s = 0.0f;
  for (int k = (int)threadIdx.x; k < nblocks; k += 256) s += lpart[k];
  red[threadIdx.x] = s;
  __syncthreads();
  for (int o = 128; o > 0; o >>= 1) {
    if ((int)threadIdx.x < o) red[threadIdx.x] += red[threadIdx.x + o];
    __syncthreads();
  }
  if (threadIdx.x == 0) {
    unsigned ne = *nev;
    out[0] = (ne > 0u) ? (-red[0] / (float)ne) : 0.0f;
  }
}

__global__ void k_zero_out(float* out) {
  if (threadIdx.x == 0 && blockIdx.x == 0) out[0] = 0.0f;
}

// ---------------------------------------------------------------------------
extern "C" void kernel_launch(void* const* d_in, const int* in_sizes, int n_in,
                              void* d_out, int out_size, void* d_ws, size_t ws_size,
                              hipStream_t stream) {
  const float* logr  = (const float*)d_in[0];
  const float* times = (const float*)d_in[1];
  const int*   cens  = (const int*)d_in[2];
  int n = in_sizes[0];

  const size_t M4 = (size_t)MBUCK * 4u;
  uint8_t* W = (uint8_t*)d_ws;
  unsigned* cntoff = (unsigned*)(W);            // counts -> (in place) suffix offsets
  unsigned* pos    = (unsigned*)(W + 1 * M4);   // scatter cursors (zeroed)
  float*    ST     = (float*)   (W + 2 * M4);   // bucket sums -> (in place) suffix sums
  unsigned* sidx   = (unsigned*)(W + 3 * M4);   // scattered original indices
  void*     bpart  = (void*)    (W + 4 * M4);                 // 2048 scan partials
  float*    lpart  = (float*)   (W + 4 * M4 + 16384);         // loss partials
  unsigned* nev    = (unsigned*)(W + 4 * M4 + 16384 + 262144);

  const size_t need = 4 * M4 + 16384 + 262144 + 64;
  if (ws_size < need || n <= 0 || (unsigned)n > MBUCK) {
    k_zero_out<<<1, 1, 0, stream>>>((float*)d_out);
    return;
  }

  // zero counts + scatter cursors + event counter (deterministic init each call)
  (void)hipMemsetAsync(W, 0, 2 * M4, stream);
  (void)hipMemsetAsync(nev, 0, sizeof(unsigned), stream);

  // 1) histogram (TDM-staged streaming reads)
  k_count<<<(n + TDM_TILE - 1) / TDM_TILE, 256, 0, stream>>>(times, cntoff, n);

  // 2) suffix-exclusive scan of counts -> counting-sort offsets (in place)
  k_scan_ph1<unsigned><<<SCAN_BLOCKS, SCAN_TPB, 0, stream>>>(cntoff, (unsigned*)bpart);
  k_scan_ph2<unsigned><<<1, SCAN_TPB, 0, stream>>>((unsigned*)bpart);
  k_scan_ph3<unsigned><<<SCAN_BLOCKS, SCAN_TPB, 0, stream>>>(cntoff, (const unsigned*)bpart);

  // 3) scatter indices into bucket segments
  k_scatter<<<(n + 255) / 256, 256, 0, stream>>>(times, cntoff, pos, sidx, n);

  // 4) per-bucket exp-sums (deterministic order)
  k_bucketsum<<<(int)(MBUCK / 256u), 256, 0, stream>>>(cntoff, sidx, logr, ST, n);

  // 5) suffix-exclusive scan of bucket sums -> at-risk base (in place)
  k_scan_ph1<float><<<SCAN_BLOCKS, SCAN_TPB, 0, stream>>>(ST, (float*)bpart);
  k_scan_ph2<float><<<1, SCAN_TPB, 0, stream>>>((float*)bpart);
  k_scan_ph3<float><<<SCAN_BLOCKS, SCAN_TPB, 0, stream>>>(ST, (const float*)bpart);

  // 6) per-event loss + 7) final reduction
  int lblocks = (n + LOSS_TPB * LOSS_ITEMS - 1) / (LOSS_TPB * LOSS_ITEMS);
  k_loss<<<lblocks, LOSS_TPB, 0, stream>>>(logr, times, cens, cntoff, sidx, ST, lpart, nev, n);
  k_final<<<1, 256, 0, stream>>>(lpart, lblocks, nev, (float*)d_out);
}